// DCNv3_Native_44117904064682
// MI455X (gfx1250) — compile-verified
//
#include <hip/hip_runtime.h>
#include <cmath>

// ---------------------------------------------------------------------------
// DCNv3 (offset-conv -> bilinear deformable sampling -> 576-reduce GEMM ->
// BN + SiLU) for MI455X / gfx1250.
//
// The op is gather/L2-bound (~85M bilinear loads vs. 2.7 GFLOP of GEMM), so
// the GEMM uses V_WMMA_F32_16X16X4_F32: full fp32, exact vs. the reference,
// and far faster than the gather front-end feeding it. This revision strips
// integer-division index math out of the hot phases (per-kk reusable gather
// indices/weights; single-row epilogue with immediate-offset stores).
// ---------------------------------------------------------------------------

typedef __attribute__((ext_vector_type(2))) float v2f;
typedef __attribute__((ext_vector_type(8))) float v8f;

constexpr int HH   = 96;
constexpr int WW   = 96;
constexpr int HWSZ = HH * WW;          // 9216
constexpr int C1   = 64;
constexpr int C2   = 64;
constexpr int KK9  = 9;                // 3x3 taps
constexpr int NBATCH = 4;
constexpr int RED  = C1 * KK9;         // 576 reduce length
constexpr int LDA  = RED + 4;          // 580: even (8B-aligned b64), bank skew

// ---------------------------------------------------------------------------
// Kernel 1: offset/mask conv  om[b][oc][ho][wo], oc in [0,27)
// ---------------------------------------------------------------------------
__global__ __launch_bounds__(256) void om_conv_kernel(
    const float* __restrict__ x, const float* __restrict__ w_om,
    const float* __restrict__ b_om, float* __restrict__ om)
{
  int idx = blockIdx.x * 256 + threadIdx.x;
  if (idx >= NBATCH * 27 * HWSZ) return;
  int wo = idx % WW;  int t = idx / WW;
  int ho = t % HH;    t /= HH;
  int oc = t % 27;    int b = t / 27;

  float acc = b_om[oc];
  const float* xb = x    + (size_t)b  * C1 * HWSZ;
  const float* wb = w_om + (size_t)oc * C1 * 9;
  #pragma unroll 1
  for (int c = 0; c < C1; ++c) {
    const float* xp = xb + c * HWSZ;
    const float* wp = wb + c * 9;
    #pragma unroll
    for (int ki = 0; ki < 3; ++ki) {
      int yy = ho - 1 + ki;
      if (yy < 0 || yy >= HH) continue;
      #pragma unroll
      for (int kj = 0; kj < 3; ++kj) {
        int xx = wo - 1 + kj;
        if (xx < 0 || xx >= WW) continue;
        acc = fmaf(xp[yy * WW + xx], wp[ki * 3 + kj], acc);
      }
    }
  }
  om[idx] = acc;
}

// ---------------------------------------------------------------------------
// Kernel 2: fused sampling + WMMA GEMM + BN + SiLU.
// One block = 128 threads = 4 wave32, owns 16 pixels (one output-row segment,
// since base % 16 == 0 and 96 % 16 == 0); wave w computes N-tile [w*16, w*16+16).
// ---------------------------------------------------------------------------
__global__ __launch_bounds__(128) void dcn_wmma_kernel(
    const float* __restrict__ x,      const float* __restrict__ om,
    const float* __restrict__ w_conv, const float* __restrict__ gamma,
    const float* __restrict__ beta,   const float* __restrict__ run_mean,
    const float* __restrict__ run_var, float* __restrict__ out)
{
  __shared__ float ldsA[16 * LDA];               // 16 pixels x 576 (padded)
  __shared__ float s_ly[144], s_lx[144], s_m[144];
  __shared__ int   s_y0[144], s_x0[144];

  const int tid  = threadIdx.x;
  const int base = blockIdx.x * 16;              // first global pixel of tile
  const int b    = base / HWSZ;                  // uniform over block
  const int loc  = base % HWSZ;
  const int ho   = loc / WW;                     // uniform: tile is one row
  const int wo0  = loc % WW;                     // wo of pixel p is wo0 + p

  // ---- Phase A: per-(pixel, tap) sampling parameters --------------------
  for (int e = tid; e < 144; e += 128) {
    int p  = e / KK9, kk = e % KK9;
    int wo = wo0 + p;
    const float* omp = om + (((size_t)b * 27) * HH + ho) * WW + wo;
    float dy = omp[(size_t)(2 * kk)     * HWSZ];   // off channel 2k
    float dx = omp[(size_t)(2 * kk + 1) * HWSZ];   // off channel 2k+1
    float mm = omp[(size_t)(18 + kk)    * HWSZ];   // mask logits
    float py = (float)(ho - 1 + kk / 3) + dy;
    float px = (float)(wo - 1 + kk % 3) + dx;
    float y0f = floorf(py), x0f = floorf(px);
    s_y0[e] = (int)y0f;  s_x0[e] = (int)x0f;
    s_ly[e] = py - y0f;  s_lx[e] = px - x0f;
    s_m[e]  = 1.0f / (1.0f + expf(-mm));
  }
  __syncthreads();

  // ---- Phase B: bilinear-sample the 16x576 A tile into LDS --------------
  // Thread owns pixel p = tid>>3, channel phase sub = tid&7. Per tap kk the
  // four clamped gather indices and validity/mask-folded weights are computed
  // once, then reused across 8 channels: no integer division in the hot loop.
  {
    const int p   = tid >> 3;
    const int sub = tid & 7;
    const float* xb = x + ((size_t)b * C1 + sub) * HWSZ;  // channel c = sub+8i
    float* arow = ldsA + p * LDA;
    #pragma unroll 1
    for (int kk = 0; kk < KK9; ++kk) {
      int e = p * KK9 + kk;
      int y0 = s_y0[e], x0 = s_x0[e];
      float ly = s_ly[e], lx = s_lx[e], m = s_m[e];
      int y1 = y0 + 1, x1 = x0 + 1;
      // clamp for safe addressing; fold validity into weights (ref: clip*valid)
      int yc0 = min(max(y0, 0), HH - 1), yc1 = min(max(y1, 0), HH - 1);
      int xc0 = min(max(x0, 0), WW - 1), xc1 = min(max(x1, 0), WW - 1);
      float vy0 = (y0 >= 0 && y0 < HH) ? 1.0f : 0.0f;
      float vy1 = (y1 >= 0 && y1 < HH) ? 1.0f : 0.0f;
      float vx0 = (x0 >= 0 && x0 < WW) ? 1.0f : 0.0f;
      float vx1 = (x1 >= 0 && x1 < WW) ? 1.0f : 0.0f;
      float hy = 1.0f - ly, hx = 1.0f - lx;
      float w00 = hy * hx * vy0 * vx0 * m;
      float w01 = hy * lx * vy0 * vx1 * m;
      float w10 = ly * hx * vy1 * vx0 * m;
      float w11 = ly * lx * vy1 * vx1 * m;
      int i00 = yc0 * WW + xc0, i01 = yc0 * WW + xc1;
      int i10 = yc1 * WW + xc0, i11 = yc1 * WW + xc1;
      const float* xp = xb;
      int aoff = sub * KK9 + kk;
      #pragma unroll
      for (int i = 0; i < 8; ++i) {               // channels sub, sub+8, ...
        float s = w00 * xp[i00] + w01 * xp[i01]
                + w10 * xp[i10] + w11 * xp[i11];
        arow[aoff] = s;
        xp   += 8 * HWSZ;
        aoff += 8 * KK9;
      }
    }
  }
  __syncthreads();

  // ---- Phase C: 16x16x576 GEMM with V_WMMA_F32_16X16X4_F32 --------------
  const int wave = tid >> 5;
  const int lane = tid & 31;
  const int half = lane >> 4;        // 0: K pair {0,1}, 1: K pair {2,3}
  const int lrow = lane & 15;        // A row (pixel) / B col (out channel)
  const int n    = wave * 16 + lrow; // this lane's output channel

  const float* ap = ldsA   + lrow * LDA        + 2 * half;  // A[m][k0+2h..]
  const float* bp = w_conv + (size_t)n * RED   + 2 * half;  // B[k0+2h..][n]
  __builtin_prefetch(bp, 0, 1);

  v8f acc = {};
  #pragma unroll 4
  for (int k0 = 0; k0 < RED; k0 += 4) {
    v2f a  = *(const v2f*)(ap + k0);    // ds_load_2addr_b64 fragments
    v2f bb = *(const v2f*)(bp + k0);    // global_load_b64 fragments
    // D = A(16x4) * B(4x16) + C, fp32 throughout
    acc = __builtin_amdgcn_wmma_f32_16x16x4_f32(
        /*neg_a=*/false, a, /*neg_b=*/false, bb,
        /*c_mod=*/(short)0, acc, /*reuse_a=*/false, /*reuse_b=*/false);
  }

  // ---- Epilogue: BN + SiLU, one row => immediate-offset stores ----------
  float sc = gamma[n] * rsqrtf(run_var[n] + 1e-5f);
  float sh = beta[n] - run_mean[n] * sc;
  // C/D layout: element v of this lane is pixel m = v + 8*half, column n.
  float* op = out + ((((size_t)b * C2 + n) * HH + ho) * WW + wo0) + 8 * half;
  #pragma unroll
  for (int v = 0; v < 8; ++v) {
    float y = acc[v] * sc + sh;
    op[v] = y / (1.0f + expf(-y));     // y * sigmoid(y)
  }
}

// ---------------------------------------------------------------------------
extern "C" void kernel_launch(void* const* d_in, const int* in_sizes, int n_in,
                              void* d_out, int out_size, void* d_ws, size_t ws_size,
                              hipStream_t stream) {
  const float* x        = (const float*)d_in[0];
  const float* w_om     = (const float*)d_in[1];
  const float* b_om     = (const float*)d_in[2];
  const float* w_conv   = (const float*)d_in[3];
  const float* gamma    = (const float*)d_in[4];
  const float* beta     = (const float*)d_in[5];
  const float* run_mean = (const float*)d_in[6];
  const float* run_var  = (const float*)d_in[7];
  float* out = (float*)d_out;
  float* om  = (float*)d_ws;                 // 4*27*9216 floats ~ 4 MB scratch

  const int total1 = NBATCH * 27 * HWSZ;     // 995,328 offset/mask outputs
  om_conv_kernel<<<(total1 + 255) / 256, 256, 0, stream>>>(x, w_om, b_om, om);

  const int nblocks = (NBATCH * HWSZ) / 16;  // 2304 pixel tiles
  dcn_wmma_kernel<<<nblocks, 128, 0, stream>>>(
      x, om, w_conv, gamma, beta, run_mean, run_var, out);
}